// tgate_topk_55679956025633
// MI455X (gfx1250) — compile-verified
//
#include <hip/hip_runtime.h>
#include <math.h>

// ---------------------------------------------------------------------------
// MoE top-2 gating head for MI455X (gfx1250), wave32 + V_WMMA_F32_16X16X4_F32.
// out[b,s] = sum_{e in top2(logits)} softmax2(logits)_e * sigmoid(x . We_e)
//   logits = x . Wc^T   (bc/be biases added in epilogue)
// Memory-bound (134 MB of x @ 23.3 TB/s ~ 5.8us floor); fp32 WMMA keeps the
// dot products at full precision so the top-2 selection matches the fp32 ref.
// ---------------------------------------------------------------------------

typedef __attribute__((ext_vector_type(2))) float v2f;
typedef __attribute__((ext_vector_type(4))) float v4f;
typedef __attribute__((ext_vector_type(8))) float v8f;

#define DIMS   1024
#define NEXP   8
#define NCOL   16              // 8 classifier cols + 8 expert cols
#define NTOK   (8 * 4096)      // B * S
#define TILE_M 16              // tokens per wave tile
#define WAVES  8               // waves per block
#define KC     64              // K chunk staged in LDS
#define XPAD   68              // padded LDS row stride (floats): banks 4r+c, conflict-free
#define WPAD   68
#define RPAD   17              // result staging stride

__global__ __launch_bounds__(256) void moe_tgate_kernel(
    const float* __restrict__ x,   // [NTOK, DIMS]
    const float* __restrict__ Wc,  // [NEXP, DIMS]
    const float* __restrict__ bc,  // [NEXP]
    const float* __restrict__ We,  // [DIMS, NEXP]
    const float* __restrict__ be,  // [NEXP]
    float* __restrict__ out)       // [NTOK]
{
    __shared__ float lds_x[WAVES][TILE_M * XPAD];  // per-wave x slab (also reused for results)
    __shared__ float lds_w[NCOL * WPAD];           // weight chunk, stored transposed: [n][d]

    const int tid  = threadIdx.x;
    const int lane = tid & 31;
    const int wave = tid >> 5;
    const int hl   = lane >> 4;   // half-wave select (0 / 1)
    const int ll   = lane & 15;

    const long tokBase = ((long)blockIdx.x * WAVES + wave) * TILE_M;
    const float* xw = x + tokBase * DIMS;

    v8f acc = {0.f, 0.f, 0.f, 0.f, 0.f, 0.f, 0.f, 0.f};

    for (int kc = 0; kc < DIMS; kc += KC) {
        __syncthreads();  // protect lds_w (and x slabs) from previous-iteration readers

        // --- stage weight chunk W_all[kc..kc+63][0..15] as lds_w[n][d] (block-cooperative)
        for (int i = tid; i < NCOL * KC; i += 256) {
            int n = i >> 6;          // / KC
            int d = i & (KC - 1);
            float w = (n < NEXP) ? Wc[n * DIMS + kc + d]
                                 : We[(kc + d) * NEXP + (n - NEXP)];
            lds_w[n * WPAD + d] = w;
        }

        // --- stage x slab 16 x KC (per-wave), coalesced b128 loads
        #pragma unroll
        for (int i = 0; i < 8; ++i) {
            int row = 2 * i + hl;            // two rows per iteration across half-waves
            int col = ll * 4;
            v4f v = *(const v4f*)(xw + row * DIMS + kc + col);
            *(v4f*)(&lds_x[wave][row * XPAD + col]) = v;
        }
        __syncthreads();

        // --- 16 WMMA steps: A 16x4 (fp32), B 4x16 (fp32), C 16x16 fp32
        const float* xl = &lds_x[wave][0];
        #pragma unroll
        for (int ks = 0; ks < KC / 4; ++ks) {
            int k4 = ks * 4 + 2 * hl;
            // A lane layout: row = ll, K-pair {k4, k4+1}
            v2f a = *(const v2f*)(xl + ll * XPAD + k4);
            // B lane layout: col = ll, K = {k4, k4+1}
            v2f b = *(const v2f*)(lds_w + ll * WPAD + k4);
            acc = __builtin_amdgcn_wmma_f32_16x16x4_f32(
                false, a, false, b, (short)0, acc, false, false);
        }
    }

    // --- stage accumulator tile to LDS: VGPR r, lane l -> token r + 8*hl, col ll
    __syncthreads();
    float* res = &lds_x[wave][0];
    #pragma unroll
    for (int r = 0; r < 8; ++r) {
        int m = r + 8 * hl;
        res[m * RPAD + ll] = acc[r];
    }
    __syncthreads();

    // --- epilogue: lanes 0..15 each finish one token scalarly
    if (lane < 16) {
        const float* rm = res + lane * RPAD;
        float lg[NEXP], pe[NEXP];
        #pragma unroll
        for (int e = 0; e < NEXP; ++e) {
            lg[e] = rm[e] + bc[e];
            pe[e] = rm[NEXP + e] + be[e];
        }
        // top-2 over logits (first-occurrence wins on ties, matches lax.top_k)
        int i1 = 0; float m1 = lg[0];
        #pragma unroll
        for (int e = 1; e < NEXP; ++e)
            if (lg[e] > m1) { m1 = lg[e]; i1 = e; }
        int i2 = (i1 == 0) ? 1 : 0; float m2 = lg[i2];
        #pragma unroll
        for (int e = 0; e < NEXP; ++e)
            if (e != i1 && lg[e] > m2) { m2 = lg[e]; i2 = e; }
        // softmax over (m1, m2); m1 >= m2 so exp argument <= 0 (stable)
        float e2  = expf(m2 - m1);
        float inv = 1.f / (1.f + e2);
        float g1  = inv;
        float g2  = e2 * inv;
        float s1 = 1.f / (1.f + expf(-pe[i1]));
        float s2 = 1.f / (1.f + expf(-pe[i2]));
        out[tokBase + lane] = g1 * s1 + g2 * s2;
    }
}

extern "C" void kernel_launch(void* const* d_in, const int* in_sizes, int n_in,
                              void* d_out, int out_size, void* d_ws, size_t ws_size,
                              hipStream_t stream) {
    (void)in_sizes; (void)n_in; (void)out_size; (void)d_ws; (void)ws_size;
    const float* x  = (const float*)d_in[0];
    const float* Wc = (const float*)d_in[1];
    const float* bc = (const float*)d_in[2];
    const float* We = (const float*)d_in[3];
    const float* be = (const float*)d_in[4];
    float* out = (float*)d_out;

    dim3 grid(NTOK / (WAVES * TILE_M));   // 256 blocks
    dim3 block(256);                      // 8 waves, one 16-token tile each
    moe_tgate_kernel<<<grid, block, 0, stream>>>(x, Wc, bc, We, be, out);
}